// NonMaximumSuppression_16484084482949
// MI455X (gfx1250) — compile-verified
//
#include <hip/hip_runtime.h>
#include <hip/hip_bf16.h>

#define NMS_B 8
#define NMS_N 4096
#define NMS_OUT 256
#define NTILE (NMS_N / 16)          // 256 column tiles of 16
#define WORDS_PER_ROW (NMS_N / 32)  // 128 u32 words per mask row

typedef float v2f __attribute__((ext_vector_type(2)));
typedef float v8f __attribute__((ext_vector_type(8)));

// -------------------- Phase 1: per-batch sort + box prep --------------------
__global__ __launch_bounds__(1024) void nms_prep_sort(const float* __restrict__ in,
                                                      float4* __restrict__ boxes_s,
                                                      float* __restrict__ area_s,
                                                      int* __restrict__ order) {
  __shared__ float key[NMS_N];
  __shared__ int sidx[NMS_N];
  const int b = blockIdx.x;
  const int tid = threadIdx.x;
  const float* bin = in + (size_t)b * NMS_N * 5;

  for (int i = tid; i < NMS_N; i += 1024) {
    key[i] = bin[(size_t)i * 5 + 0];
    sidx[i] = i;
  }
  __syncthreads();

  // Bitonic sort, descending by key, ties broken by ascending original index.
  for (int k = 2; k <= NMS_N; k <<= 1) {
    for (int j = k >> 1; j > 0; j >>= 1) {
      for (int t = tid; t < NMS_N; t += 1024) {
        int ixj = t ^ j;
        if (ixj > t) {
          float av = key[t], bv = key[ixj];
          int ia = sidx[t], ib = sidx[ixj];
          bool dir = ((t & k) == 0);  // true -> descending segment
          bool a_before_b = (av > bv) || (av == bv && ia < ib);
          bool b_before_a = (bv > av) || (bv == av && ib < ia);
          bool swap = dir ? b_before_a : a_before_b;
          if (swap) {
            key[t] = bv; key[ixj] = av;
            sidx[t] = ib; sidx[ixj] = ia;
          }
        }
      }
      __syncthreads();
    }
  }

  // Emit sorted corners / areas / order. Note w//2 == floor(w/2) in reference.
  for (int p = tid; p < NMS_N; p += 1024) {
    int orig = sidx[p];
    const float* r = bin + (size_t)orig * 5;
    float x = r[1], y = r[2], w = r[3], h = r[4];
    float ws = floorf(w * 0.5f);
    float hs = floorf(h * 0.5f);
    float4 bx;
    bx.x = x - ws; bx.y = y - hs; bx.z = x + ws; bx.w = y + hs;
    boxes_s[(size_t)b * NMS_N + p] = bx;
    area_s[(size_t)b * NMS_N + p] = (2.0f * ws) * (2.0f * hs);
    order[(size_t)b * NMS_N + p] = orig;
  }
}

// -------------------- Phase 2: 16x16 IoU mask tiles (one wave32 per tile) ---
__global__ __launch_bounds__(256) void nms_mask(const float4* __restrict__ boxes_s,
                                                const float* __restrict__ area_s,
                                                unsigned short* __restrict__ mask16) {
  const int lane = threadIdx.x & 31;
  const int tile = blockIdx.x * (256 >> 5) + (threadIdx.x >> 5);
  const int b = tile >> 16;          // tile / (NTILE*NTILE)
  const int rem = tile & 0xFFFF;
  const int tr = rem >> 8;           // row tile
  const int tc = rem & 0xFF;         // col tile
  unsigned short* mrow = mask16 + (size_t)b * NMS_N * NTILE;

  if (tc < tr) {  // strictly lower-triangular tile: all bits are j<i -> zero
    if (lane < 16) mrow[(size_t)(tr * 16 + lane) * NTILE + tc] = 0;
    return;
  }

  const int rowBase = tr * 16, colBase = tc * 16;
  const float4* bb = boxes_s + (size_t)b * NMS_N;
  const float* aa = area_s + (size_t)b * NMS_N;
  const int n = lane & 15;

  float4 bj = bb[colBase + n];
  float aj = aa[colBase + n];
  float ai = aa[rowBase + n];

  // areaSum[m][n] = area_i[m] + area_j[n] as a rank-2 outer product on the
  // matrix pipe: A(16x4) = [a_i, 1, 0, 0], B(4x16) = [1^T; a_j^T; 0; 0].
  // A layout: lanes 0-15 hold K=0,1; lanes 16-31 hold K=2,3 (zeros).
  // B layout: VGPR0 lanes 0-15 = row K=0, lanes 16-31 = row K=1.
  const bool lo = lane < 16;
  v2f A, Bm;
  A.x = lo ? ai : 0.0f;  A.y = lo ? 1.0f : 0.0f;
  Bm.x = lo ? 1.0f : aj; Bm.y = 0.0f;
  v8f c = {};
  c = __builtin_amdgcn_wmma_f32_16x16x4_f32(false, A, false, Bm, (short)0, c,
                                            false, false);

  // Prefetch the next column tile's boxes into cache while WMMA retires.
  __builtin_prefetch((const void*)(bb + colBase + 16), 0, 1);

  const int half = lane >> 4;  // C layout: lanes16-31 hold rows M=8..15
  const int jglob = colBase + n;
#pragma unroll
  for (int m = 0; m < 8; ++m) {
    int gi = rowBase + m + (half << 3);
    float4 bi = bb[gi];
    float ix1 = fmaxf(bi.x, bj.x);
    float iy1 = fmaxf(bi.y, bj.y);
    float ix2 = fminf(bi.z, bj.z);
    float iy2 = fminf(bi.w, bj.w);
    float inter = fmaxf(ix2 - ix1, 0.0f) * fmaxf(iy2 - iy1, 0.0f);
    // iou > 0.5  <=>  inter > 0.5*(sum - inter)  <=>  3*inter > sum
    bool pred = (3.0f * inter > c[m]) && (jglob > gi);
    unsigned int bal = __builtin_amdgcn_ballot_w32(pred);
    if (lane == 0)
      mrow[(size_t)(rowBase + m) * NTILE + tc] = (unsigned short)(bal & 0xFFFF);
    if (lane == 16)
      mrow[(size_t)(rowBase + 8 + m) * NTILE + tc] = (unsigned short)(bal >> 16);
  }
}

// -------------------- Phase 3: greedy scan (one wave32 per batch) -----------
__global__ __launch_bounds__(32) void nms_scan(const float* __restrict__ in,
                                               const int* __restrict__ order,
                                               const unsigned short* __restrict__ mask16,
                                               float* __restrict__ out) {
  const int b = blockIdx.x;
  const int lane = threadIdx.x;
  const float* bin = in + (size_t)b * NMS_N * 5;
  const unsigned int* mbase =
      (const unsigned int*)(mask16 + (size_t)b * NMS_N * NTILE);

  // 4096-bit removed-vector: lane L owns words 4L..4L+3.
  unsigned int r0 = 0, r1 = 0, r2 = 0, r3 = 0;
  int cnt = 0;
  int i = 0;
  while (i < NMS_N && cnt < NMS_OUT) {
    int w = i >> 5;
    int srcLane = w >> 2, slot = w & 3;
    unsigned int word = (slot == 0) ? r0 : (slot == 1) ? r1 : (slot == 2) ? r2 : r3;
    word = (unsigned int)__builtin_amdgcn_readlane((int)word, srcLane);
    unsigned int avail = ~word & (0xFFFFFFFFu << (i & 31));
    if (!avail) { i = (w + 1) << 5; continue; }
    i = (w << 5) + __builtin_ctz(avail);  // next unsuppressed candidate

    // keep sorted position i
    int orig = order[(size_t)b * NMS_N + i];
    if (lane < 4)
      out[((size_t)b * NMS_OUT + cnt) * 4 + lane] = bin[(size_t)orig * 5 + 1 + lane];
    cnt++;

    // OR in the mask row of i (bits j<=i are guaranteed zero)
    const unsigned int* rowp = mbase + (size_t)i * WORDS_PER_ROW + lane * 4;
    uint4 m = *(const uint4*)rowp;
    r0 |= m.x; r1 |= m.y; r2 |= m.z; r3 |= m.w;
    i++;
  }

  // Reference pads with random boxes (non-reproducible); pad deterministically.
  for (int k = cnt; k < NMS_OUT; ++k)
    if (lane < 4)
      out[((size_t)b * NMS_OUT + k) * 4 + lane] = bin[1 + lane];
}

extern "C" void kernel_launch(void* const* d_in, const int* in_sizes, int n_in,
                              void* d_out, int out_size, void* d_ws, size_t ws_size,
                              hipStream_t stream) {
  const float* in = (const float*)d_in[0];
  float* out = (float*)d_out;

  // Workspace carve-up (~16.75 MB total)
  float4* boxes_s = (float4*)d_ws;                               // 512 KB
  float* area_s = (float*)(boxes_s + (size_t)NMS_B * NMS_N);     // 128 KB
  int* order = (int*)(area_s + (size_t)NMS_B * NMS_N);           // 128 KB
  unsigned short* mask16 = (unsigned short*)(order + (size_t)NMS_B * NMS_N); // 16 MB

  nms_prep_sort<<<NMS_B, 1024, 0, stream>>>(in, boxes_s, area_s, order);

  // 8 * 256 * 256 tiles, 8 wave32 per 256-thread block
  const int totalTiles = NMS_B * NTILE * NTILE;
  nms_mask<<<totalTiles / 8, 256, 0, stream>>>(boxes_s, area_s, mask16);

  nms_scan<<<NMS_B, 32, 0, stream>>>(in, order, mask16, out);
}